// CurvatureTotal_42812234006549
// MI455X (gfx1250) — compile-verified
//
#include <hip/hip_runtime.h>
#include <cstdint>

// Problem constants (fixed by the reference: z is (8,1,1024,1024) fp32, h=30, eps=1e-12)
#define NIMG  8
#define H_IMG 1024
#define W_IMG 1024

// Tiling: 128 x 32 outputs per 256-thread (8-wave) block.
#define TX 128
#define TY 32
#define ROWS (TY + 2)     // tile rows incl. 1-row halo top/bottom
#define PITCH 136         // floats per LDS row: col 3 = left halo, cols 4..131 interior, col 132 = right halo
                          // interior starts at byte offset 16 -> all B128 LDS accesses are 16B aligned

// Convert a generic pointer to a __shared__ object into its 32-bit LDS byte address.
__device__ __forceinline__ unsigned lds_off(const void* p) {
    return (unsigned)(unsigned long long)(__attribute__((address_space(3))) const void*)p;
}

// CDNA5 async global->LDS copies (per-lane, tracked with ASYNCcnt).
__device__ __forceinline__ void async_load_b128(unsigned lds, const float* g) {
    asm volatile("global_load_async_to_lds_b128 %0, %1, off"
                 :: "v"(lds), "v"(g)
                 : "memory");
}
__device__ __forceinline__ void async_load_b32(unsigned lds, const float* g) {
    asm volatile("global_load_async_to_lds_b32 %0, %1, off"
                 :: "v"(lds), "v"(g)
                 : "memory");
}
__device__ __forceinline__ void wait_asynccnt0() {
    asm volatile("s_wait_asynccnt 0" ::: "memory");
}

// Load 6 consecutive taps starting at LDS column (xbase+3): one aligned float4 + 2 scalars.
__device__ __forceinline__ void load6(const float* rowp, float a[6]) {
    a[0] = rowp[3];                                           // left neighbor (ds_load_b32)
    const float4 v = *reinterpret_cast<const float4*>(rowp + 4);  // aligned ds_load_b128
    a[1] = v.x; a[2] = v.y; a[3] = v.z; a[4] = v.w;
    a[5] = rowp[8];                                           // right neighbor (ds_load_b32)
}

__global__ __launch_bounds__(256)
void CurvatureTotal_kernel(const float* __restrict__ z, float* __restrict__ out) {
    __shared__ __align__(16) float tile[ROWS * PITCH];

    const int x0  = blockIdx.x * TX;
    const int y0  = blockIdx.y * TY;
    const int n   = blockIdx.z;
    const int tid = threadIdx.x;

    const float* img = z + (size_t)n * H_IMG * W_IMG;

    // ---- Stage tile (with halo) into LDS via async copies -------------------
    // Interior: ROWS rows x (TX/4) 16-byte segments, row-clamped (replicate pad).
    for (int i = tid; i < ROWS * (TX / 4); i += 256) {
        const int r   = i / (TX / 4);
        const int seg = i % (TX / 4);
        int gy = y0 - 1 + r;
        gy = gy < 0 ? 0 : (gy > H_IMG - 1 ? H_IMG - 1 : gy);
        const float* gp = img + (size_t)gy * W_IMG + x0 + seg * 4;
        async_load_b128(lds_off(&tile[r * PITCH + 4 + seg * 4]), gp);
    }
    // Halo columns (x0-1 and x0+TX), clamped to the image (replicate pad).
    for (int i = tid; i < ROWS * 2; i += 256) {
        const int r    = i >> 1;
        const int side = i & 1;
        int gy = y0 - 1 + r;
        gy = gy < 0 ? 0 : (gy > H_IMG - 1 ? H_IMG - 1 : gy);
        const int gx = side ? (x0 + TX <= W_IMG - 1 ? x0 + TX : W_IMG - 1)
                            : (x0 - 1 >= 0 ? x0 - 1 : 0);
        const int lc = side ? (4 + TX) : 3;
        async_load_b32(lds_off(&tile[r * PITCH + lc]), img + (size_t)gy * W_IMG + gx);
    }
    wait_asynccnt0();       // each wave's own async copies complete
    __syncthreads();        // all waves' copies visible to the workgroup

    // ---- Compute: each thread produces 4 consecutive x per row, 8 rows/pass --
    const float h      = 30.0f;
    const float inv2h  = 1.0f / (2.0f * h);
    const float invh2  = 1.0f / (h * h);
    const float inv4h2 = 0.25f * invh2;
    const float eps    = 1e-12f;

    const int lane  = tid & 31;     // 32 lanes cover TX = 128 floats (4 each)
    const int rg    = tid >> 5;     // 0..7: row group
    const int xbase = lane * 4;     // x within tile, multiple of 4
    const int gxb   = x0 + xbase;

    // Per-sub-element zero-pad masks (conv2d padding=1 semantics); only edge tiles differ from 1.
    float wl[4], wr[4];
#pragma unroll
    for (int j = 0; j < 4; ++j) {
        const int gx = gxb + j;
        wl[j] = (gx == 0)         ? 0.0f : 1.0f;
        wr[j] = (gx == W_IMG - 1) ? 0.0f : 1.0f;
    }

#pragma unroll
    for (int k = 0; k < 4; ++k) {
        const int r  = rg + 8 * k;      // 0..31
        const int gy = y0 + r;
        const float wu = (gy == 0)         ? 0.0f : 1.0f;
        const float wd = (gy == H_IMG - 1) ? 0.0f : 1.0f;

        float a0[6], a1[6], a2[6];
        load6(&tile[(r    ) * PITCH + xbase], a0);  // row gy-1 (clamped in LDS)
        load6(&tile[(r + 1) * PITCH + xbase], a1);  // row gy
        load6(&tile[(r + 2) * PITCH + xbase], a2);  // row gy+1 (clamped in LDS)

        float4 res;
        float* rp = &res.x;
#pragma unroll
        for (int j = 0; j < 4; ++j) {
            const float zl  = a1[j],     zc  = a1[j + 1], zr  = a1[j + 2];
            const float zu  = a0[j + 1], zd  = a2[j + 1];
            const float zul = a0[j],     zur = a0[j + 2];
            const float zdl = a2[j],     zdr = a2[j + 2];

            // First derivatives: replicate padding == clamped LDS values directly.
            const float p = (zr - zl) * inv2h;
            const float q = (zd - zu) * inv2h;
            // Second derivatives: zero padding == mask out off-image taps.
            const float rxx = (wr[j] * zr - 2.0f * zc + wl[j] * zl) * invh2;
            const float tyy = (wd * zd    - 2.0f * zc + wu * zu   ) * invh2;
            const float s   = (wu * wl[j] * zul - wu * wr[j] * zur
                             - wd * wl[j] * zdl + wd * wr[j] * zdr) * inv4h2;

            const float p2 = p * p, q2 = q * q, pq = p * q;
            const float dg = p2 + q2 + eps;          // denom_g
            const float gp = 1.0f + p2 + q2 + eps;   // one_plus_g + eps
            const float A  = -(p2 * rxx + 2.0f * pq * s + q2 * tyy);   // Cp numerator
            const float B  = -(q2 * rxx - 2.0f * pq * s + p2 * tyy);   // Ct numerator
            // Cp*Ct = A*B / (dg^2 * gp^1.5 * gp^0.5) = A*B / (dg*dg*gp*gp)  (no sqrt needed)
            rp[j] = (A * B) / (dg * dg * gp * gp);
        }
        // Coalesced 16B store: 32 lanes x 16B = 512B contiguous per wave-row.
        *reinterpret_cast<float4*>(out + ((size_t)n * H_IMG + gy) * W_IMG + gxb) = res;
    }
}

extern "C" void kernel_launch(void* const* d_in, const int* in_sizes, int n_in,
                              void* d_out, int out_size, void* d_ws, size_t ws_size,
                              hipStream_t stream) {
    (void)in_sizes; (void)n_in; (void)out_size; (void)d_ws; (void)ws_size;
    const float* zin = (const float*)d_in[0];
    float* outp = (float*)d_out;
    dim3 grid(W_IMG / TX, H_IMG / TY, NIMG);   // 8 x 32 x 8 = 2048 blocks
    CurvatureTotal_kernel<<<grid, 256, 0, stream>>>(zin, outp);
}